// ENCTCcudaLoss_11965778886978
// MI455X (gfx1250) — compile-verified
//
#include <hip/hip_runtime.h>
#include <hip/hip_bf16.h>
#include <cmath>

// Problem constants (match reference setup_inputs exactly)
#define Tt   256
#define Bb   64
#define Cc   6625
#define Ss   32
#define LL   65              // 2*S+1 extended label length
#define NEGV (-1e30f)
#define TINYF (1e-30f)
#define HRATE 0.2f

// Async prefetch pipeline: DEPTH rows in flight, 3 async instr per row per wave
#define DEPTH 8
#define NSLOT (DEPTH + 1)
#define ROWP  96             // ring row stride (floats): 32 lanes * 3 states, no OOB

// ---------------------------------------------------------------------------
// logaddexp matching jnp.logaddexp: max + log1p(exp(min-max))
__device__ __forceinline__ float lae(float a, float b) {
  float m = fmaxf(a, b);
  float d = fminf(a, b) - m;          // <= 0 always (0 for ties, incl NEG,NEG)
  return m + log1pf(expf(d));
}

// gfx1250 ASYNCcnt path: per-lane async global->LDS dword copy.
__device__ __forceinline__ void async_gather_b32(unsigned lds_byte_off,
                                                 const float* gaddr) {
  asm volatile("global_load_async_to_lds_b32 %0, %1, off"
               :: "v"(lds_byte_off), "v"(gaddr)
               : "memory");
}

// 24 outstanding steady-state; <=21 retires the oldest row's 3 loads (in-order).
__device__ __forceinline__ void wait_async_le21() {
  asm volatile("s_wait_asynccnt 0x15" ::: "memory");
}

// ---------------------------------------------------------------------------
// One single wave32 per batch sample. Lane l owns states 3l..3l+2 entirely in
// registers; neighbor states come from __shfl_up. Zero barriers in the T-loop.
__global__ __launch_bounds__(32)
void enctc_dp_kernel(const float* __restrict__ log_probs,   // [T,B,C]
                     const int*   __restrict__ targets,     // [B,S]
                     const int*   __restrict__ inlen,       // [B]
                     const int*   __restrict__ tlen,        // [B]
                     float*       __restrict__ loss_out)    // [B]
{
  const int b = blockIdx.x;
  const int l = threadIdx.x;

  __shared__ float ring[NSLOT][ROWP];   // async gather landing zone
  __shared__ float finA[LL + 1];
  __shared__ float finH[LL + 1];

  // Per-state extended label + skip flag (clamped so all 32 lanes are safe).
  int  e[3];  bool skip[3];  bool valid[3];
#pragma unroll
  for (int j = 0; j < 3; ++j) {
    int s = 3 * l + j;
    valid[j] = (s < LL);
    e[j] = 0; skip[j] = false;
    if (valid[j] && (s & 1)) {
      int jj = s >> 1;                       // 0..31 in-range for valid odd s
      e[j]    = targets[b * Ss + jj];
      skip[j] = (s >= 3) && (e[j] != targets[b * Ss + jj - 1]);
    }
  }

  // Prologue: prefetch rows 0..DEPTH-1. All 32 lanes issue all 3 gathers so
  // EXEC is uniform and ASYNCcnt advances exactly 3 per row per wave.
#pragma unroll
  for (int k = 0; k < DEPTH; ++k) {
#pragma unroll
    for (int j = 0; j < 3; ++j) {
      const float* gp = log_probs + ((size_t)k * Bb + b) * Cc + e[j];
      async_gather_b32((unsigned)(size_t)&ring[k][3 * l + j], gp);
    }
  }

  const int mylen = inlen[b];
  float a0 = NEGV, a1 = NEGV, a2 = NEGV;     // alpha for states 3l,3l+1,3l+2
  float h0 = NEGV, h1 = NEGV, h2 = NEGV;     // lH
  float aF0 = NEGV, aF1 = NEGV, aF2 = NEGV;  // snapshot at t == inlen-1
  float hF0 = NEGV, hF1 = NEGV, hF2 = NEGV;

  int slot = 0, pslot = DEPTH % NSLOT;

  for (int t = 0; t < Tt; ++t) {
    wait_async_le21();                       // row t resident in ring[slot]
    float lp0 = ring[slot][3 * l + 0];
    float lp1 = ring[slot][3 * l + 1];
    float lp2 = ring[slot][3 * l + 2];

    if (t == 0) {
      bool i0 = (l == 0);                    // states 0 and 1 live on lane 0
      a0 = i0 ? lp0 : NEGV;
      h0 = i0 ? lp0 + logf(fmaxf(-lp0, TINYF)) : NEGV;
      a1 = i0 ? lp1 : NEGV;
      h1 = i0 ? lp1 + logf(fmaxf(-lp1, TINYF)) : NEGV;
      a2 = NEGV; h2 = NEGV;
    } else {
      // previous-step neighbors: prev lane's a1/a2/h1/h2 via wave32 shuffle
      float ua1 = __shfl_up(a1, 1), ua2 = __shfl_up(a2, 1);
      float uh1 = __shfl_up(h1, 1), uh2 = __shfl_up(h2, 1);
      bool  edge = (l == 0);
      float p_a1 = edge ? NEGV : ua1, p_a2 = edge ? NEGV : ua2;
      float p_h1 = edge ? NEGV : uh1, p_h2 = edge ? NEGV : uh2;

      // s=3l   : s-1 = prev.a2, s-2 = prev.a1
      // s=3l+1 : s-1 = own a0,  s-2 = prev.a2
      // s=3l+2 : s-1 = own a1,  s-2 = own a0
      float A0 = lae(lae(a0, p_a2), skip[0] ? p_a1 : NEGV);
      float H0 = lae(lae(h0, p_h2), skip[0] ? p_h1 : NEGV);
      float A1 = lae(lae(a1, a0),  skip[1] ? p_a2 : NEGV);
      float H1 = lae(lae(h1, h0),  skip[1] ? p_h2 : NEGV);
      float A2 = lae(lae(a2, a1),  skip[2] ? a0  : NEGV);
      float H2 = lae(lae(h2, h1),  skip[2] ? h0  : NEGV);

      float ln0 = logf(fmaxf(-lp0, TINYF));
      float ln1 = logf(fmaxf(-lp1, TINYF));
      float ln2 = logf(fmaxf(-lp2, TINYF));

      a0 = lp0 + A0;  h0 = lp0 + lae(H0, A0 + ln0);
      a1 = lp1 + A1;  h1 = lp1 + lae(H1, A1 + ln1);
      a2 = lp2 + A2;  h2 = lp2 + lae(H2, A2 + ln2);
      if (!valid[0]) { a0 = NEGV; h0 = NEGV; }
      if (!valid[1]) { a1 = NEGV; h1 = NEGV; }
      if (!valid[2]) { a2 = NEGV; h2 = NEGV; }
    }

    if (t == mylen - 1) {
      aF0 = a0; aF1 = a1; aF2 = a2;
      hF0 = h0; hF1 = h1; hF2 = h2;
    }

    // Prefetch row t+DEPTH (clamped dup near the tail keeps ASYNCcnt
    // saturated so the single immediate wait value is valid every step).
    int tr = t + DEPTH; if (tr > Tt - 1) tr = Tt - 1;
#pragma unroll
    for (int j = 0; j < 3; ++j) {
      const float* gp = log_probs + ((size_t)tr * Bb + b) * Cc + e[j];
      async_gather_b32((unsigned)(size_t)&ring[pslot][3 * l + j], gp);
    }

    slot  = (slot  + 1 == NSLOT) ? 0 : slot  + 1;
    pslot = (pslot + 1 == NSLOT) ? 0 : pslot + 1;
  }

  if (valid[0]) { finA[3 * l + 0] = aF0; finH[3 * l + 0] = hF0; }
  if (valid[1]) { finA[3 * l + 1] = aF1; finH[3 * l + 1] = hF1; }
  if (valid[2]) { finA[3 * l + 2] = aF2; finH[3 * l + 2] = hF2; }
  __syncthreads();                            // single-wave WG: trivial

  if (l == 0) {
    int   tl   = tlen[b];
    int   i1   = 2 * tl - 1, i2 = 2 * tl;     // tl>=16 so i1>=31, i2<=64
    float logP = lae(finA[i1], finA[i2]);
    float lHf  = lae(finH[i1], finH[i2]);
    float nll  = -logP;
    float ent  = expf(lHf - logP) + logP;
    float loss = (1.0f - HRATE) * nll - HRATE * ent;
    if (!__builtin_isfinite(loss)) loss = 0.0f;   // zero_infinity
    loss_out[b] = loss / (float)tl;               // torch per-length scaling
  }
}

// ---------------------------------------------------------------------------
__global__ void enctc_reduce_kernel(const float* __restrict__ lossb,
                                    float* __restrict__ out)
{
  __shared__ float sh[Bb];
  int i = threadIdx.x;
  sh[i] = lossb[i];
  __syncthreads();
  if (i == 0) {
    float acc = 0.0f;
    for (int j = 0; j < Bb; ++j) acc += sh[j];
    out[0] = acc / (float)Bb;                     // mean over batch
  }
}

// ---------------------------------------------------------------------------
extern "C" void kernel_launch(void* const* d_in, const int* in_sizes, int n_in,
                              void* d_out, int out_size, void* d_ws, size_t ws_size,
                              hipStream_t stream) {
  (void)in_sizes; (void)n_in; (void)out_size; (void)ws_size;
  const float* log_probs = (const float*)d_in[0];   // [T,B,C] f32
  const int*   targets   = (const int*)  d_in[1];   // [B,S]   i32
  const int*   inlen     = (const int*)  d_in[2];   // [B]     i32
  const int*   tlen      = (const int*)  d_in[3];   // [B]     i32
  float*       lossb     = (float*)d_ws;            // [B] scratch
  float*       out       = (float*)d_out;           // scalar

  hipLaunchKernelGGL(enctc_dp_kernel, dim3(Bb), dim3(32), 0, stream,
                     log_probs, targets, inlen, tlen, lossb);
  hipLaunchKernelGGL(enctc_reduce_kernel, dim3(1), dim3(Bb), 0, stream,
                     lossb, out);
}